// RetinaFaceDecoder_72198400246419
// MI455X (gfx1250) — compile-verified
//
#include <hip/hip_runtime.h>
#include <hip/hip_bf16.h>
#include <math.h>

// RetinaFace decode, CDNA5 (gfx1250).
// Streaming kernel: 172 B/anchor of HBM traffic, ~7.4 us roofline at 23.3 TB/s.
// Uses the CDNA5 async-to-LDS pipeline (global_load_async_to_lds_b128 +
// s_wait_asynccnt) with double-buffered 256-anchor tiles.

#define TILE 256
#define ROWF 20   // padded LDS row: 16 data floats + 4 pad (80B, 16B-aligned, breaks 8-way bank conflict)

#if __has_builtin(__builtin_amdgcn_global_load_async_to_lds_b128)
#define HAVE_ASYNC_B128 1
#endif
#if __has_builtin(__builtin_amdgcn_s_wait_asynccnt)
#define HAVE_WAIT_ASYNC 1
#endif

typedef int v4i __attribute__((ext_vector_type(4)));
typedef __attribute__((address_space(1))) v4i GV4;   // global int4
typedef __attribute__((address_space(3))) v4i LV4;   // LDS int4

__device__ __forceinline__ void async_copy_b128(const float* __restrict__ gsrc,
                                                float* ldst) {
#if defined(HAVE_ASYNC_B128)
  __builtin_amdgcn_global_load_async_to_lds_b128(
      (GV4*)(void*)gsrc, (LV4*)(void*)ldst, /*offset=*/0, /*cpol=*/0);
#else
  unsigned lofs = (unsigned)(unsigned long long)ldst;  // low 32 bits = LDS byte offset
  asm volatile("global_load_async_to_lds_b128 %0, %1, off"
               :: "v"(lofs), "v"(gsrc) : "memory");
#endif
}

__device__ __forceinline__ void wait_async_le5() {
#if defined(HAVE_WAIT_ASYNC)
  __builtin_amdgcn_s_wait_asynccnt(5);
#else
  asm volatile("s_wait_asynccnt 0x5" ::: "memory");
#endif
}

__device__ __forceinline__ void wait_async_le0() {
#if defined(HAVE_WAIT_ASYNC)
  __builtin_amdgcn_s_wait_asynccnt(0);
#else
  asm volatile("s_wait_asynccnt 0x0" ::: "memory");
#endif
}

// Stage one tile (256 anchors) into LDS buffers. Issues EXACTLY 5 wave-level
// async ops (4x input B128 + 1x priors B128) per thread with indices clamped
// (not predicated) so ASYNCcnt bookkeeping stays exact and EXEC stays full.
__device__ __forceinline__ void stage_tile(long long tile, long long n,
                                           const float* __restrict__ input,
                                           const float* __restrict__ priors,
                                           float* in_s, float* pr_s) {
  const int t = threadIdx.x;
  const long long chunkBase = tile * (long long)(TILE * 4);  // 16B chunks of input
  const long long maxChunk  = n * 4 - 1;
#pragma unroll
  for (int k = 0; k < 4; ++k) {
    const int idx = t + k * TILE;          // 0..1023, contiguous 16B chunks
    long long g = chunkBase + idx;
    g = (g > maxChunk) ? maxChunk : g;     // branchless tail clamp (no EXEC change)
    const int a = idx >> 2, c = idx & 3;   // anchor slot / chunk-in-row
    async_copy_b128(input + g * 4, in_s + (a * ROWF + c * 4));
  }
  long long p = tile * TILE + t;
  p = (p > n - 1) ? (n - 1) : p;
  async_copy_b128(priors + p * 4, pr_s + t * 4);
}

__global__ __launch_bounds__(TILE) void retinaface_decode_kernel(
    const float* __restrict__ input,   // (N,16) flat
    const float* __restrict__ thrp,    // (1,)
    const float* __restrict__ priors,  // (N,4)
    const float* __restrict__ varp,    // (2,)
    float* __restrict__ out,           // 23*N floats, concat of 5 outputs
    long long n, long long numTiles) {
  __shared__ float in_s[2][TILE * ROWF];  // 2 x 20KB
  __shared__ float pr_s[2][TILE * 4];     // 2 x 4KB

  const float thr = thrp[0];
  const float v0  = varp[0];
  const float v1  = varp[1];

  float* __restrict__ ob = out;            // boxes_m   : 4N
  float* __restrict__ oc = out + 4 * n;    // conf_m    : N
  float* __restrict__ op = out + 5 * n;    // pred_m    : N (always 0 by construction)
  float* __restrict__ od = out + 6 * n;    // detections: 16N
  float* __restrict__ ok = out + 22 * n;   // keep      : N

  const int t = threadIdx.x;
  long long tile = blockIdx.x;
  const long long stride = gridDim.x;
  int buf = 0;

  if (tile < numTiles)
    stage_tile(tile, n, input, priors, &in_s[buf][0], &pr_s[buf][0]);

  for (; tile < numTiles; tile += stride) {
    const long long nextT = tile + stride;
    if (nextT < numTiles) {
      stage_tile(nextT, n, input, priors, &in_s[buf ^ 1][0], &pr_s[buf ^ 1][0]);
      wait_async_le5();   // 5 outstanding = next tile; in-order decrement => current tile done
    } else {
      wait_async_le0();
    }
    __syncthreads();      // current tile's LDS visible to all 8 waves

    const long long i = tile * TILE + t;
    if (i < n) {
      const float4* rv = (const float4*)&in_s[buf][t * ROWF];
      const float4 A = rv[0];                       // loc
      const float4 B = rv[1];                       // c0,c1,lm0,lm1
      const float4 C = rv[2];                       // lm2..lm5
      const float4 D = rv[3];                       // lm6..lm9
      const float4 P = *(const float4*)&pr_s[buf][t * 4];  // cx,cy,pw,ph

      // boxes
      const float bx  = P.x + (A.x * v0) * P.z;
      const float by  = P.y + (A.y * v0) * P.w;
      const float bw  = P.z * expf(A.z * v1);
      const float bh  = P.w * expf(A.w * v1);
      const float tlx = bx - bw * 0.5f;
      const float tly = by - bh * 0.5f;
      const float brx = tlx + bw;
      const float bry = tly + bh;

      // 2-class softmax: conf = 1/(1+exp(-|c1-c0|)), pred = (c1 > c0)
      const float dlt  = B.y - B.x;
      const int   pred = (dlt > 0.0f) ? 1 : 0;
      const float conf = 1.0f / (1.0f + expf(-fabsf(dlt)));

      const bool  keep = (pred > 0) && (conf > thr);
      const float mf   = keep ? 1.0f : 0.0f;
      const float pm   = keep ? (float)(pred - 1) : 0.0f;  // provably 0

      // landmarks
      const float l0 = P.x + (B.z * v0) * P.z;
      const float l1 = P.y + (B.w * v0) * P.w;
      const float l2 = P.x + (C.x * v0) * P.z;
      const float l3 = P.y + (C.y * v0) * P.w;
      const float l4 = P.x + (C.z * v0) * P.z;
      const float l5 = P.y + (C.w * v0) * P.w;
      const float l6 = P.x + (D.x * v0) * P.z;
      const float l7 = P.y + (D.y * v0) * P.w;
      const float l8 = P.x + (D.z * v0) * P.z;
      const float l9 = P.y + (D.w * v0) * P.w;

      const float4 bx4 = make_float4(tlx * mf, tly * mf, brx * mf, bry * mf);
      *(float4*)&ob[4 * i] = bx4;                   // global_store_b128, coalesced
      oc[i] = conf * mf;
      op[i] = pm;
      ok[i] = mf;
      float4* dp = (float4*)&od[16 * i];
      dp[0] = bx4;
      dp[1] = make_float4(conf * mf, pm, l0 * mf, l1 * mf);
      dp[2] = make_float4(l2 * mf, l3 * mf, l4 * mf, l5 * mf);
      dp[3] = make_float4(l6 * mf, l7 * mf, l8 * mf, l9 * mf);
    }
    __syncthreads();      // all waves done reading buf before it is restaged
    buf ^= 1;
  }
}

extern "C" void kernel_launch(void* const* d_in, const int* in_sizes, int n_in,
                              void* d_out, int out_size, void* d_ws, size_t ws_size,
                              hipStream_t stream) {
  const float* input  = (const float*)d_in[0];  // (1,N,16) float32
  const float* thr    = (const float*)d_in[1];  // (1,)     float32
  const float* priors = (const float*)d_in[2];  // (N,4)    float32
  const float* var    = (const float*)d_in[3];  // (2,)     float32
  float* out = (float*)d_out;

  const long long n        = (long long)in_sizes[0] / 16;
  const long long numTiles = (n + TILE - 1) / TILE;
  const int grid = (int)((numTiles < 1024) ? numTiles : 1024);

  retinaface_decode_kernel<<<grid, TILE, 0, stream>>>(input, thr, priors, var,
                                                      out, n, numTiles);
}